// ParallelHeadNet_11768210391044
// MI455X (gfx1250) — compile-verified
//
#include <hip/hip_runtime.h>
#include <math.h>

// ---------------------------------------------------------------------------
// ParallelHeadNet for MI455X (gfx1250, wave32).
// Bandwidth-bound pipeline (~168 MB streamed, ~82 MFLOP => pure streaming):
//   k1 : sub = qw @ x^T             -- real GEMM -> V_WMMA_F32_16X16X4_F32
//   k2a: z1 = w1[h] @ sub[h]        -- per-head GEMV, NT b128 (16.8 MB), 256 blks
//   k2b: h1 = softplus(GN(z1))      -- per-head epilogue
//   k3a: z  = w2[s,h] @ h1[h]       -- NT b128 stream (134 MB), 1024 blks
//   k3b: zsm = softmax(GN(z))       -- per-(s,h) epilogue (in-place buffer)
//   k4 : scalar = rowsum(ws)        -- wave-per-row NT b128 (16.8 MB)
//   k5 : gate/sigmoid/outputs       -- fused epilogue
// GEMV/epilogue split maximizes in-flight HBM requests (8192 waves on k3a).
// ---------------------------------------------------------------------------

typedef float v2f __attribute__((ext_vector_type(2)));
typedef float v4f __attribute__((ext_vector_type(4)));
typedef float v8f __attribute__((ext_vector_type(8)));

#define N_INP   128
#define N_HID   512
#define N_OUT   256
#define N_HEADS 64
#define N_SM    4
#define EPS_GN  1e-5f
#define TINY    1e-14f

// ---------------- wave32 / block reductions --------------------------------

__device__ __forceinline__ float wave_sum(float v) {
#pragma unroll
  for (int off = 16; off > 0; off >>= 1) v += __shfl_down(v, off, 32);
  return v;  // full sum valid in lane 0
}

__device__ __forceinline__ float wave_max(float v) {
#pragma unroll
  for (int off = 16; off > 0; off >>= 1) v = fmaxf(v, __shfl_down(v, off, 32));
  return v;
}

__device__ __forceinline__ float block_sum(float v, float* s_red) {
  const int wave = threadIdx.x >> 5, lane = threadIdx.x & 31;
  const int nw = blockDim.x >> 5;
  float w = wave_sum(v);
  __syncthreads();                 // protect s_red reuse across calls
  if (lane == 0) s_red[wave] = w;
  __syncthreads();
  float tot = 0.f;
  for (int i = 0; i < nw; ++i) tot += s_red[i];
  return tot;
}

__device__ __forceinline__ float block_max(float v, float* s_red) {
  const int wave = threadIdx.x >> 5, lane = threadIdx.x & 31;
  const int nw = blockDim.x >> 5;
  float w = wave_max(v);
  __syncthreads();
  if (lane == 0) s_red[wave] = w;
  __syncthreads();
  float tot = -INFINITY;
  for (int i = 0; i < nw; ++i) tot = fmaxf(tot, s_red[i]);
  return tot;
}

// ---------------- k1: sub[h,r] = sum_c qw[h,c] * x[r,c]  (WMMA f32) --------
// 32 blocks (4 M-tiles x 8 N-tiles), 32 threads = one wave, EXEC all-ones.
// A (16x4, ISA 7.12.2): lanes 0-15 hold {K=0,K=1} of row M=lane,
//                       lanes 16-31 hold {K=2,K=3} of row M=lane-16.
// B (4x16) symmetric; D (16x16 f32): VGPR v -> M = v + 8*(lane>=16), N=lane&15.

__global__ __launch_bounds__(32) void k1_sub_wmma(const float* __restrict__ x,
                                                  const float* __restrict__ qw,
                                                  float* __restrict__ sub) {
  const int tm = (blockIdx.x >> 3) << 4;   // head-tile base
  const int tn = (blockIdx.x & 7) << 4;    // r-tile base
  const int lane = threadIdx.x;
  const int lo = lane & 15;
  const int hi = lane >> 4;                // 0 or 1
  const float* arow = qw + (tm + lo) * N_INP;   // qw row (contiguous in c)
  const float* brow = x  + (tn + lo) * N_INP;   // B[k][n] = x[n][k]
  v8f acc = {};
#pragma unroll 4
  for (int k0 = 0; k0 < N_INP; k0 += 4) {
    const int kb = k0 + (hi << 1);
    v2f a, b;
    a.x = arow[kb]; a.y = arow[kb + 1];
    b.x = brow[kb]; b.y = brow[kb + 1];
    acc = __builtin_amdgcn_wmma_f32_16x16x4_f32(
        /*neg_a=*/false, a, /*neg_b=*/false, b,
        /*c_mod=*/(short)0, acc, /*reuse_a=*/false, /*reuse_b=*/false);
  }
#pragma unroll
  for (int v = 0; v < 8; ++v)
    sub[(tm + v + (hi << 3)) * N_INP + (tn + lo)] = acc[v];
}

// ---------------- k2a: z1[h,r] = w1[h,r,:] . sub[h,:] ----------------------
// 256 blocks (4 per head), 8 waves, wave-per-row, contiguous float4 loads.

__global__ __launch_bounds__(256) void k2a_gemv(const float* __restrict__ w1,
                                                const float* __restrict__ sub,
                                                float* __restrict__ z1) {
  const int h = blockIdx.x >> 2;
  const int q = blockIdx.x & 3;            // quarter of the 512 rows
  __shared__ float s_sub[N_INP];
  const int t = threadIdx.x;
  if (t < N_INP) s_sub[t] = sub[h * N_INP + t];
  __syncthreads();

  const int wave = t >> 5, lane = t & 31;
  const int ib = lane << 2;
  const float s0 = s_sub[ib], s1v = s_sub[ib + 1],
              s2v = s_sub[ib + 2], s3v = s_sub[ib + 3];
  const float* w1h = w1 + (size_t)h * N_HID * N_INP;
  const int r0 = q << 7;                   // q*128
  for (int r = r0 + wave; r < r0 + 128; r += 8) {
    v4f w = __builtin_nontemporal_load((const v4f*)(w1h + r * N_INP + ib));
    float p = w.x * s0 + w.y * s1v + w.z * s2v + w.w * s3v;
    p = wave_sum(p);
    if (lane == 0) z1[h * N_HID + r] = p;
  }
}

// ---------------- k2b: h1 = softplus(GN(z1)) per head ----------------------

__global__ __launch_bounds__(256) void k2b_gn_sp(const float* __restrict__ z1,
                                                 const float* __restrict__ g1,
                                                 const float* __restrict__ b1,
                                                 float* __restrict__ h1) {
  const int h = blockIdx.x;
  __shared__ float s_red[8];
  const int t = threadIdx.x;
  const float va = z1[h * N_HID + t];
  const float vb = z1[h * N_HID + t + 256];
  const float sum = block_sum(va + vb, s_red);
  const float sq  = block_sum(va * va + vb * vb, s_red);
  const float mu   = sum * (1.f / N_HID);
  const float rstd = rsqrtf(sq * (1.f / N_HID) - mu * mu + EPS_GN);

  float zn = (va - mu) * rstd * g1[h * N_HID + t] + b1[h * N_HID + t];
  h1[h * N_HID + t] = (zn > 20.f) ? zn : log1pf(expf(zn));
  zn = (vb - mu) * rstd * g1[h * N_HID + t + 256] + b1[h * N_HID + t + 256];
  h1[h * N_HID + t + 256] = (zn > 20.f) ? zn : log1pf(expf(zn));
}

// ---------------- k3a: z[s,h,r] = w2[s,h,r,:] . h1[h,:] --------------------
// 1024 blocks (4 per (s,h)) x 8 waves = 8192 waves streaming the 134 MB w2
// with NT b128 loads + prefetch; wave-per-row dot products.

__global__ __launch_bounds__(256) void k3a_gemv(const float* __restrict__ w2,
                                                const float* __restrict__ h1,
                                                float* __restrict__ zb) {
  const int sh = blockIdx.x >> 2;          // s*64 + h
  const int q  = blockIdx.x & 3;
  const int h  = sh & (N_HEADS - 1);
  __shared__ float s_h1[N_HID];
  const int t = threadIdx.x;
  s_h1[t]       = h1[h * N_HID + t];
  s_h1[t + 256] = h1[h * N_HID + t + 256];
  __syncthreads();

  const int wave = t >> 5, lane = t & 31;
  const float* w2b = w2 + (size_t)sh * N_OUT * N_HID;
  const int r0 = q << 6;                   // q*64
  for (int r = r0 + wave; r < r0 + 64; r += 8) {
    const float* row = w2b + r * N_HID;
    if (r + 8 < r0 + 64)
      __builtin_prefetch(row + 8 * N_HID + (lane << 4), 0, 0); // next row, 64B/lane
    float p = 0.f;
#pragma unroll
    for (int j = 0; j < 4; ++j) {
      const int ib = (j << 7) + (lane << 2);   // contiguous 512B per wave-load
      v4f w = __builtin_nontemporal_load((const v4f*)(row + ib));
      p += w.x * s_h1[ib] + w.y * s_h1[ib + 1] +
           w.z * s_h1[ib + 2] + w.w * s_h1[ib + 3];
    }
    p = wave_sum(p);
    if (lane == 0) zb[sh * N_OUT + r] = p;
  }
}

// ---------------- k3b: zsm = softmax(GN(z)) per (s,h), in-place safe -------

__global__ __launch_bounds__(256) void k3b_gn_sm(const float* __restrict__ zb,
                                                 const float* __restrict__ g2,
                                                 const float* __restrict__ b2,
                                                 float* __restrict__ zsm) {
  const int sh = blockIdx.x;
  __shared__ float s_red[8];
  const int t = threadIdx.x;
  const float val = zb[sh * N_OUT + t];    // read before any write -> may alias
  const float sum = block_sum(val, s_red);
  const float sq  = block_sum(val * val, s_red);
  const float mu   = sum * (1.f / N_OUT);
  const float rstd = rsqrtf(sq * (1.f / N_OUT) - mu * mu + EPS_GN);
  const float zn = (val - mu) * rstd * g2[sh * N_OUT + t] + b2[sh * N_OUT + t];

  const float m = block_max(zn, s_red);
  const float e = expf(zn - m);
  const float S = block_sum(e, s_red);
  zsm[(size_t)sh * N_OUT + t] = e / S;
}

// ---------------- k4: scalar[o,h] = sum_j ws[h,o,j] ------------------------
// Wave-per-row: 16384 rows of 256 floats, NT b128 streaming.

__global__ __launch_bounds__(256) void k4_scalar(const float* __restrict__ wsw,
                                                 float* __restrict__ scl) {
  const int gw = (blockIdx.x * 256 + threadIdx.x) >> 5;   // 0..16383
  const int lane = threadIdx.x & 31;
  const float* row = wsw + (size_t)gw * N_OUT;
  v4f a = __builtin_nontemporal_load((const v4f*)(row + (lane << 2)));
  v4f b = __builtin_nontemporal_load((const v4f*)(row + 128 + (lane << 2)));
  float p = a.x + a.y + a.z + a.w + b.x + b.y + b.z + b.w;
  p = wave_sum(p);
  if (lane == 0) {
    const int hh = gw >> 8, o = gw & 255;
    scl[o * N_HEADS + hh] = p;   // transposed: (n_out, n_heads)
  }
}

// ---------------- k5: gate + outputs ---------------------------------------
// One block per head; thread o handles output element (o, h).

__global__ __launch_bounds__(256) void k5_out(const float* __restrict__ zsm,
                                              const float* __restrict__ scl,
                                              const float* __restrict__ last,
                                              const float* __restrict__ woo,
                                              float* __restrict__ out) {
  const int h = blockIdx.x;
  __shared__ float s_red[8];
  const int o = threadIdx.x;

  float lp = 1.f;                                  // prod over heads
#pragma unroll 8
  for (int hh = 0; hh < N_HEADS; ++hh) lp *= last[o * N_HEADS + hh];

  float osm = 0.f;                                 // sum over s branches
#pragma unroll
  for (int s = 0; s < N_SM; ++s)
    osm += zsm[((size_t)(s * N_HEADS + h)) * N_OUT + o];

  const float gp = woo[h * 2 * N_OUT + o] * lp +
                   woo[h * 2 * N_OUT + N_OUT + o] * osm;
  const float logit = block_sum(gp, s_red);        // broadcast to all threads
  const float onoff = 1.f / (1.f + expf(-logit));

  const float v1 = onoff * osm;
  out[o * N_HEADS + h] = fmaxf(v1, TINY);          // clip(.., TINY, None)

  const float v2 = v1 * scl[o * N_HEADS + h];
  out[N_OUT * N_HEADS + o * N_HEADS + h] = (fabsf(v2) <= TINY) ? TINY : v2;
}

// ---------------------------------------------------------------------------

extern "C" void kernel_launch(void* const* d_in, const int* in_sizes, int n_in,
                              void* d_out, int out_size, void* d_ws, size_t ws_size,
                              hipStream_t stream) {
  const float* x    = (const float*)d_in[0];   // (128,128)
  const float* last = (const float*)d_in[1];   // (256,64)
  const float* qw   = (const float*)d_in[2];   // (64,128)
  const float* w1   = (const float*)d_in[3];   // (64,512,128)
  const float* g1   = (const float*)d_in[4];   // (64,512)
  const float* b1   = (const float*)d_in[5];
  const float* w2   = (const float*)d_in[6];   // (4,64,256,512)
  const float* g2   = (const float*)d_in[7];   // (4,64,256)
  const float* b2   = (const float*)d_in[8];
  const float* wsw  = (const float*)d_in[9];   // (64,256,256)
  const float* woo  = (const float*)d_in[10];  // (64,512)
  float* out = (float*)d_out;                  // 2 x (256,64) concatenated
  float* ws  = (float*)d_ws;

  // workspace (floats): sub 8192 | z1 32768 | h1 32768 | zbuf 65536 | scl 16384
  float* sub  = ws;
  float* z1   = ws + 8192;
  float* h1   = ws + 8192 + 32768;
  float* zbuf = ws + 8192 + 32768 + 32768;          // softmax written in-place
  float* scl  = ws + 8192 + 32768 + 32768 + 65536;  // total 622,592 bytes

  k1_sub_wmma<<<32,   32, 0, stream>>>(x, qw, sub);
  k2a_gemv   <<<256, 256, 0, stream>>>(w1, sub, z1);
  k2b_gn_sp  <<<64,  256, 0, stream>>>(z1, g1, b1, h1);
  k3a_gemv   <<<1024,256, 0, stream>>>(w2, h1, zbuf);
  k3b_gn_sm  <<<256, 256, 0, stream>>>(zbuf, g2, b2, zbuf);
  k4_scalar  <<<2048,256, 0, stream>>>(wsw, scl);
  k5_out     <<<64,  256, 0, stream>>>(zbuf, scl, last, woo, out);
}